// Actor_15324443312913
// MI455X (gfx1250) — compile-verified
//
#include <hip/hip_runtime.h>
#include <hip/hip_bf16.h>
#include <math.h>

typedef __attribute__((ext_vector_type(2))) float v2f;
typedef __attribute__((ext_vector_type(8))) float v8f;

namespace {

constexpr int BMt = 128;   // block M tile
constexpr int BNt = 128;   // block N tile
constexpr int KTt = 32;    // K step
constexpr int LDA = 36;    // padded A stride (dwords): conflict-free b64 frag loads
constexpr int LDB = 136;   // padded B stride (dwords): conflict-free b32 frag loads

struct GemmP {
  const float* X;
  const float* W;
  const float* bias;
  float* Y;
  int K;
  int N;
  int xChunkW;               // feature-chunk width of X (concat chunk or K)
  long long xChunkStride;    // element stride between feature chunks
  long long xRowStride;      // element stride between batch rows
  long long xGroupStride;    // element stride between groups
  long long wGroupStride;    // K*N per group
  long long biasGroupStride; // N per group
  long long outGroupStride;
  long long outRowStride;
  int act;                   // 0 = linear, 1 = relu, 2 = tanh
};

// Async copy: 16B per lane, cache -> LDS, tracked by ASYNCcnt.
// INST_OFFSET is added to BOTH the LDS and the global address (ISA 10.x),
// so one LDS base + one global base serves 4 consecutive 16B chunks.
#define ASYNC_B128(lds_u32, gaddr_u64)                                          \
  asm volatile("global_load_async_to_lds_b128 %0, %1, off"                      \
               :: "v"(lds_u32), "v"(gaddr_u64) : "memory")
#define ASYNC_B128_OFF(lds_u32, gaddr_u64, OFF)                                 \
  asm volatile("global_load_async_to_lds_b128 %0, %1, off offset:" #OFF        \
               :: "v"(lds_u32), "v"(gaddr_u64) : "memory")

__global__ __launch_bounds__(256)
void fnn_gemm(GemmP p) {
  __shared__ float As[2][BMt * LDA];   // 2 x 18.0 KB
  __shared__ float Bs[2][KTt * LDB];   // 2 x 17.0 KB

  const int g   = blockIdx.z;
  const int bm0 = blockIdx.x * BMt;
  const int bn0 = blockIdx.y * BNt;

  const int tid  = threadIdx.x;
  const int wave = tid >> 5;
  const int lane = tid & 31;
  const int ll   = lane & 15;
  const int lh   = lane >> 4;
  const int wm   = (wave & 3) << 5;   // 0,32,64,96
  const int wn   = (wave >> 2) << 6;  // 0,64

  v8f acc[2][4];
#pragma unroll
  for (int mi = 0; mi < 2; ++mi)
#pragma unroll
    for (int ni = 0; ni < 4; ++ni)
      acc[mi][ni] = v8f{0.f, 0.f, 0.f, 0.f, 0.f, 0.f, 0.f, 0.f};

  const float* Xg = p.X + (long long)g * p.xGroupStride;
  const float* Wg = p.W + (long long)g * p.wGroupStride;

  // staging mapping: 256 threads, 16 floats (4 x b128) each per tile half
  const int ar  = tid >> 1;         // A row 0..127
  const int ak  = (tid & 1) << 4;   // A col chunk: 0 or 16
  const int bkr = tid >> 3;         // B row 0..31
  const int bnq = (tid & 7) << 4;   // B col chunk: 0..112

  const int ng = bn0 + bnq;

  // issue async prefetch of K-tile k0 into LDS buffer `buf`
  auto stage = [&](int k0, int buf) {
    {
      const long long chunk  = k0 / p.xChunkW;              // KT never crosses a chunk
      const long long within = (long long)(k0 % p.xChunkW) + ak;
      const float* xp = Xg + chunk * p.xChunkStride
                      + (long long)(bm0 + ar) * p.xRowStride + within;
      const unsigned lA = (unsigned)(uintptr_t)&As[buf][ar * LDA + ak];
      const unsigned long long gA = (unsigned long long)(uintptr_t)xp;
      ASYNC_B128(lA, gA);
      ASYNC_B128_OFF(lA, gA, 16);
      ASYNC_B128_OFF(lA, gA, 32);
      ASYNC_B128_OFF(lA, gA, 48);
    }
    if (ng < p.N) {
      const float* wp = Wg + (long long)(k0 + bkr) * p.N + ng;
      const unsigned lB = (unsigned)(uintptr_t)&Bs[buf][bkr * LDB + bnq];
      const unsigned long long gB = (unsigned long long)(uintptr_t)wp;
      ASYNC_B128(lB, gB);
      ASYNC_B128_OFF(lB, gB, 16);
      ASYNC_B128_OFF(lB, gB, 32);
      ASYNC_B128_OFF(lB, gB, 48);
    } else {
      float4 z = {0.f, 0.f, 0.f, 0.f};
      float4* bv = (float4*)&Bs[buf][bkr * LDB + bnq];
      bv[0] = z; bv[1] = z; bv[2] = z; bv[3] = z;   // DScnt-tracked, waited by barrier
    }
  };

  int buf = 0;
  stage(0, 0);

  for (int k0 = 0; k0 < p.K; k0 += KTt) {
    asm volatile("s_wait_asynccnt 0x0" ::: "memory");  // my async writes to buf done
    __syncthreads();                                   // everyone's done + prev reads done
    if (k0 + KTt < p.K) stage(k0 + KTt, buf ^ 1);      // prefetch next while computing

    const float* Asb = As[buf];
    const float* Bsb = Bs[buf];
#pragma unroll
    for (int ks = 0; ks < KTt; ks += 4) {
      v2f a[2];
#pragma unroll
      for (int mi = 0; mi < 2; ++mi)
        a[mi] = *(const v2f*)&Asb[(wm + mi * 16 + ll) * LDA + ks + 2 * lh];
      v2f b[4];
#pragma unroll
      for (int ni = 0; ni < 4; ++ni) {
        const float* bp = &Bsb[(ks + 2 * lh) * LDB + wn + ni * 16 + ll];
        b[ni].x = bp[0];
        b[ni].y = bp[LDB];
      }
#pragma unroll
      for (int mi = 0; mi < 2; ++mi)
#pragma unroll
        for (int ni = 0; ni < 4; ++ni)
          acc[mi][ni] = __builtin_amdgcn_wmma_f32_16x16x4_f32(
              false, a[mi], false, b[ni], (short)0, acc[mi][ni], false, false);
    }
    buf ^= 1;
  }

  // ---- epilogue: bias + activation, scatter to strided output ----
  const float* biasg = p.bias + (long long)g * p.biasGroupStride;
  float* Yg = p.Y + (long long)g * p.outGroupStride;
#pragma unroll
  for (int ni = 0; ni < 4; ++ni) {
    const int col = bn0 + wn + ni * 16 + ll;
    if (col < p.N) {
      const float bv = biasg[col];
#pragma unroll
      for (int mi = 0; mi < 2; ++mi) {
#pragma unroll
        for (int r = 0; r < 8; ++r) {
          const int row = bm0 + wm + mi * 16 + lh * 8 + r;  // C/D layout: M = r + 8*lh
          float v = acc[mi][ni][r] + bv;
          if (p.act == 1) v = fmaxf(v, 0.0f);
          else if (p.act == 2) v = tanhf(v);
          Yg[(long long)row * p.outRowStride + col] = v;
        }
      }
    }
  }
}

void launch(const float* X, const float* W, const float* bias, float* Y,
            int G, int K, int N,
            int xChunkW, long long xChunkStride, long long xRowStride,
            long long xGroupStride, long long wGroupStride,
            long long biasGroupStride, long long outGroupStride,
            long long outRowStride, int act, hipStream_t s) {
  GemmP p;
  p.X = X; p.W = W; p.bias = bias; p.Y = Y;
  p.K = K; p.N = N;
  p.xChunkW = xChunkW; p.xChunkStride = xChunkStride;
  p.xRowStride = xRowStride; p.xGroupStride = xGroupStride;
  p.wGroupStride = wGroupStride; p.biasGroupStride = biasGroupStride;
  p.outGroupStride = outGroupStride; p.outRowStride = outRowStride;
  p.act = act;
  dim3 grid(4096 / BMt, (unsigned)((N + BNt - 1) / BNt), (unsigned)G);
  hipLaunchKernelGGL(fnn_gemm, grid, dim3(256), 0, s, p);
}

} // namespace

extern "C" void kernel_launch(void* const* d_in, const int* in_sizes, int n_in,
                              void* d_out, int out_size, void* d_ws, size_t ws_size,
                              hipStream_t stream) {
  (void)in_sizes; (void)n_in; (void)out_size; (void)ws_size;
  constexpr long long B = 4096, D = 128, E = 64;
  constexpr long long ROW = 85 * D;  // embeds batch-row stride

  const float* state  = (const float*)d_in[0];
  const float* enc_w1 = (const float*)d_in[1];
  const float* enc_b1 = (const float*)d_in[2];
  const float* enc_wh = (const float*)d_in[3];
  const float* enc_bh = (const float*)d_in[4];
  const float* enc_wo = (const float*)d_in[5];
  const float* enc_bo = (const float*)d_in[6];
  const float *agg_w1[3], *agg_b1[3], *agg_wh[3], *agg_bh[3], *agg_wo[3], *agg_bo[3];
  for (int li = 0; li < 3; ++li) {
    const int base = 7 + li * 6;
    agg_w1[li] = (const float*)d_in[base + 0];
    agg_b1[li] = (const float*)d_in[base + 1];
    agg_wh[li] = (const float*)d_in[base + 2];
    agg_bh[li] = (const float*)d_in[base + 3];
    agg_wo[li] = (const float*)d_in[base + 4];
    agg_bo[li] = (const float*)d_in[base + 5];
  }
  const float* head_w1 = (const float*)d_in[25];
  const float* head_b1 = (const float*)d_in[26];
  const float* head_wh = (const float*)d_in[27];
  const float* head_bh = (const float*)d_in[28];
  const float* head_wo = (const float*)d_in[29];
  const float* head_bo = (const float*)d_in[30];

  float* emb    = (float*)d_out;          // [B, 85, D]
  float* action = emb + B * 85 * D;       // [B, 32]
  float* h1 = (float*)d_ws;               // 33,554,432 floats
  float* h2 = h1 + E * B * D;             // 33,554,432 floats

  // encoders: state [B,S] broadcast to 64 groups
  launch(state, enc_w1, enc_b1, h1, 64, 256, 128,
         256, 0, 256, 0, 256 * 128, 128, B * 128, 128, 1, stream);
  launch(h1, enc_wh, enc_bh, h2, 64, 128, 128,
         128, 0, 128, B * 128, 128 * 128, 128, B * 128, 128, 1, stream);
  launch(h2, enc_wo, enc_bo, emb, 64, 128, 128,                 // -> embeds nodes 0..63
         128, 0, 128, B * 128, 128 * 128, 128, 128, ROW, 0, stream);

  // aggregator levels: input = 4 sibling embeddings concat, read directly from embeds
  const int nlev[3] = {16, 4, 1};
  const long long nodeOff[4] = {0, 64, 80, 84};
  for (int li = 0; li < 3; ++li) {
    const int n = nlev[li];
    const float* xin = emb + nodeOff[li] * D;
    launch(xin, agg_w1[li], agg_b1[li], h1, n, 512, 512,
           128, 128, ROW, 4 * 128, 512 * 512, 512, B * 512, 512, 1, stream);
    launch(h1, agg_wh[li], agg_bh[li], h2, n, 512, 512,
           512, 0, 512, B * 512, 512 * 512, 512, B * 512, 512, 1, stream);
    launch(h2, agg_wo[li], agg_bo[li], emb + nodeOff[li + 1] * D, n, 512, 128,
           512, 0, 512, B * 512, 512 * 128, 128, 128, ROW, 0, stream);
  }

  // head on the root embedding (node 84)
  const float* root = emb + 84 * D;
  launch(root, head_w1, head_b1, h1, 1, 128, 128,
         128, 0, ROW, 0, 0, 0, 0, 128, 1, stream);
  launch(h1, head_wh, head_bh, h2, 1, 128, 128,
         128, 0, 128, 0, 0, 0, 0, 128, 1, stream);
  launch(h2, head_wo, head_bo, action, 1, 128, 32,
         128, 0, 128, 0, 0, 0, 0, 32, 2, stream);
}